// MaxAttSentence_16063177687231
// MI455X (gfx1250) — compile-verified
//
#include <hip/hip_runtime.h>

// Problem constants (from reference): B=32, L=4096, D=768, S=128, M=64
#define B_ 32
#define L_ 4096
#define D_ 768
#define S_ 128
#define M_ 64

// ---- CDNA5 async global<->LDS path (gfx1250) -------------------------------
// Builtins probe-confirmed on this toolchain (round-1 diagnostics showed the
// signatures: (AS1 v4i* global, AS3 v4i* lds, imm offset, imm cpol)).
#if defined(__HIP_DEVICE_COMPILE__) && \
    __has_builtin(__builtin_amdgcn_global_load_async_to_lds_b128) && \
    __has_builtin(__builtin_amdgcn_global_store_async_from_lds_b128) && \
    __has_builtin(__builtin_amdgcn_s_wait_asynccnt)
#define USE_ASYNC_LDS 1
#else
#define USE_ASYNC_LDS 0
#endif

#if USE_ASYNC_LDS
typedef int v4i __attribute__((ext_vector_type(4)));
typedef __attribute__((address_space(1))) v4i* g_v4i_p; // global
typedef __attribute__((address_space(3))) v4i* l_v4i_p; // LDS
#endif

// ---------------------------------------------------------------------------
// Kernel 1: per-batch sentence selection. One block per batch, 128 threads.
//
// Phase A: stage attention[b, 0:4096] (16 KB) into LDS with coalesced async
//   b128 loads — 8 chunks x 128 lanes, all 8 requests in flight per lane,
//   ONE s_wait_asynccnt + barrier = a single global round trip (the previous
//   version serialized up to 32 dependent global loads per lane).
// Phase B: thread t sums its ragged span row[s0:e0) from LDS (cheap ds loads),
//   then an LDS tree reduction finds the FIRST maximum (tie-break on smaller
//   sentence index = jnp.argmax semantics). Thread 0 applies the
//   (max_sum > 0) validity rule and writes {bstart, blen} to the workspace.
// ---------------------------------------------------------------------------
__global__ __launch_bounds__(S_) void MaxAttSentence_select_kernel(
    const int* __restrict__ startends,   // [B, S, 2]
    const float* __restrict__ attention, // [B, L]
    int* __restrict__ bs)                // [B, 2] -> {bstart, blen}
{
    const int b = blockIdx.x;
    const int t = threadIdx.x; // sentence id 0..127

    __shared__ float row[L_]; // 16 KB attention row

    const float* att = attention + (size_t)b * L_;

    // ---- Phase A: coalesced staging of the attention row into LDS --------
#if USE_ASYNC_LDS
    #pragma unroll
    for (int c = 0; c < L_ / 4 / S_; ++c) {       // 8 float4 chunks per lane
        const int i4 = c * S_ + t;                // coalesced float4 index
        __builtin_amdgcn_global_load_async_to_lds_b128(
            (g_v4i_p)(att + (size_t)i4 * 4), (l_v4i_p)(row + i4 * 4), 0, 0);
    }
    __builtin_amdgcn_s_wait_asynccnt(0);          // this wave's LDS writes done
#else
    #pragma unroll
    for (int c = 0; c < L_ / 4 / S_; ++c) {
        const int i4 = c * S_ + t;
        ((float4*)row)[i4] = ((const float4*)att)[i4];
    }
#endif
    __syncthreads();                              // all waves' writes visible

    // ---- Phase B: ragged segment sums from LDS ---------------------------
    const int s0 = startends[((size_t)b * S_ + t) * 2 + 0];
    const int e0 = startends[((size_t)b * S_ + t) * 2 + 1];

    float sum = 0.0f;
    for (int i = s0; i < e0; ++i) sum += row[i];

    __shared__ float sh_sum[S_];
    __shared__ int   sh_idx[S_];
    __shared__ int   sh_s[S_];
    __shared__ int   sh_e[S_];
    sh_sum[t] = sum;
    sh_idx[t] = t;
    sh_s[t]   = s0;
    sh_e[t]   = e0;
    __syncthreads();

    // Tree reduction: max value, first (lowest) index on ties.
    for (int off = S_ / 2; off > 0; off >>= 1) {
        if (t < off) {
            const float a  = sh_sum[t];
            const float c  = sh_sum[t + off];
            const int   ia = sh_idx[t];
            const int   ic = sh_idx[t + off];
            if (c > a || (c == a && ic < ia)) {
                sh_sum[t] = c;
                sh_idx[t] = ic;
            }
        }
        __syncthreads();
    }

    if (t == 0) {
        const int   best    = sh_idx[0];
        const float max_sum = sh_sum[0];
        int bstart = 0, blen = 0;
        if (max_sum > 0.0f) {
            bstart = sh_s[best];
            blen   = sh_e[best] - sh_s[best];
        }
        bs[b * 2 + 0] = bstart;
        bs[b * 2 + 1] = blen;
    }
}

// ---------------------------------------------------------------------------
// Kernel 2: gather + pad. One block per output row (grid = 64 x 32),
// 192 threads x 16B = 3072B = one full row of D=768 floats.
// Valid rows (pos < blen) are streamed global->LDS->global with the CDNA5
// async engine (each lane owns its private 16B chunk, so only an ASYNCcnt
// wait is needed between the load and the store — no barrier). Padded rows
// are zeroed with b128 stores. The branch is uniform per block, so EXEC is
// all-ones inside the async region. No wait after the final async store:
// S_ENDPGM performs an implicit S_WAIT_IDLE.
// ---------------------------------------------------------------------------
__global__ __launch_bounds__(D_ / 4) void MaxAttSentence_gather_kernel(
    const float* __restrict__ context, // [B, L, D]
    const int* __restrict__ bs,        // [B, 2]
    float* __restrict__ out)           // [B, M, D]
{
    const int pos = blockIdx.x;  // 0..M-1
    const int b   = blockIdx.y;  // 0..B-1
    const int t   = threadIdx.x; // 0..191, one float4 each

    const int bstart = bs[b * 2 + 0];
    const int blen   = bs[b * 2 + 1];

    float* dst = out + ((size_t)b * M_ + pos) * D_ + (size_t)t * 4;

    if (pos < blen) {
        const float* src =
            context + ((size_t)b * L_ + (size_t)(bstart + pos)) * D_ + (size_t)t * 4;
#if USE_ASYNC_LDS
        __shared__ float lrow[D_];
        float* lds_elem = &lrow[t * 4];
        // global -> LDS (async, b128)
        __builtin_amdgcn_global_load_async_to_lds_b128(
            (g_v4i_p)src, (l_v4i_p)lds_elem, 0, 0);
        __builtin_amdgcn_s_wait_asynccnt(0);
        // LDS -> global (async, b128); same lane-private 16B chunk.
        // Completion is covered by S_ENDPGM's implicit wait-idle.
        __builtin_amdgcn_global_store_async_from_lds_b128(
            (g_v4i_p)dst, (l_v4i_p)lds_elem, 0, 0);
#else
        *(float4*)dst = *(const float4*)src;
#endif
    } else {
        *(float4*)dst = make_float4(0.0f, 0.0f, 0.0f, 0.0f);
    }
}

// ---------------------------------------------------------------------------
extern "C" void kernel_launch(void* const* d_in, const int* in_sizes, int n_in,
                              void* d_out, int out_size, void* d_ws, size_t ws_size,
                              hipStream_t stream) {
    const int*   startends = (const int*)d_in[0];   // [B, S, 2] int32
    const float* attention = (const float*)d_in[1]; // [B, L] f32
    const float* context   = (const float*)d_in[2]; // [B, L, D] f32
    float*       out       = (float*)d_out;         // [B, M, D] f32
    int*         bs        = (int*)d_ws;            // [B, 2] scratch (256 B)

    (void)in_sizes; (void)n_in; (void)out_size; (void)ws_size;

    MaxAttSentence_select_kernel<<<B_, S_, 0, stream>>>(startends, attention, bs);
    MaxAttSentence_gather_kernel<<<dim3(M_, B_), D_ / 4, 0, stream>>>(context, bs, out);
}